// LineGCN_30442728194391
// MI455X (gfx1250) — compile-verified
//
#include <hip/hip_runtime.h>

#define NN 100000   // nodes
#define EE 1600000  // edges
#define DD 128      // feature dim

typedef __attribute__((ext_vector_type(2))) float v2f;
typedef __attribute__((ext_vector_type(8))) float v8f;

// ---------------------------------------------------------------------------
// 1. degree histograms (f32 atomics; L2-resident, 400KB each)
// ---------------------------------------------------------------------------
__global__ void __launch_bounds__(256) deg_kernel(const int* __restrict__ src,
                                                  const int* __restrict__ dst,
                                                  float* __restrict__ degs,
                                                  float* __restrict__ degd) {
  int t = blockIdx.x * blockDim.x + threadIdx.x;
  if (t < EE) {
    atomicAdd(&degs[src[t]], 1.0f);
    atomicAdd(&degd[dst[t]], 1.0f);
  }
}

__global__ void __launch_bounds__(256) norm_kernel(const float* __restrict__ degs,
                                                   const float* __restrict__ degd,
                                                   float* __restrict__ out_norm,
                                                   float* __restrict__ in_norm) {
  int i = blockIdx.x * blockDim.x + threadIdx.x;
  if (i < NN) {
    out_norm[i] = rsqrtf(fmaxf(degs[i], 1.0f));
    in_norm[i]  = rsqrtf(fmaxf(degd[i], 1.0f));
  }
}

// ---------------------------------------------------------------------------
// 2. f32 WMMA GEMM:  C[M x 128] = (diag(rowScale) * A[M x 128]) @ B[128 x 128]
//    One wave -> 16 rows x 64 cols (4 accumulators). Block = 8 waves = 64 rows.
//    Uses V_WMMA_F32_16X16X4_F32, K-loop of 32 steps.
// ---------------------------------------------------------------------------
__global__ void __launch_bounds__(256) gemm_f32_wmma(const float* __restrict__ A,
                                                     const float* __restrict__ B,
                                                     const float* __restrict__ rowScale, // may be null
                                                     float* __restrict__ C, int M) {
  const int lane = threadIdx.x & 31;
  const int wave = threadIdx.x >> 5;
  const int half = lane >> 4;   // 0: lanes 0-15, 1: lanes 16-31
  const int lm   = lane & 15;

  const int row0 = blockIdx.x * 64 + (wave >> 1) * 16; // wave-uniform
  if (row0 >= M) return;                               // whole wave exits -> EXEC all-1s at WMMA
  const int col0 = (wave & 1) * 64;

  const int arow = row0 + lm;
  float s = 1.0f;
  if (rowScale) s = rowScale[arow];

  const float* __restrict__ Arow = A + (size_t)arow * DD;

  v8f acc0 = {}, acc1 = {}, acc2 = {}, acc3 = {};

#pragma unroll 4
  for (int k0 = 0; k0 < DD; k0 += 4) {
    const int ka = k0 + 2 * half;
    // A fragment 16x4: lane holds A[M=lm][ka], A[M=lm][ka+1]
    v2f a;
    {
      const float2 av = *(const float2*)(Arow + ka);
      a.x = av.x * s;
      a.y = av.y * s;
    }
    // B fragments 4x16: VGPR0 = row ka (lanes 0-15) / row ka+2... per-lane: rows ka, ka+1
    const float* __restrict__ B0 = B + (size_t)ka * DD + col0 + lm;
    const float* __restrict__ B1 = B0 + DD;
    v2f b0, b1, b2, b3;
    b0.x = B0[0];  b0.y = B1[0];
    b1.x = B0[16]; b1.y = B1[16];
    b2.x = B0[32]; b2.y = B1[32];
    b3.x = B0[48]; b3.y = B1[48];

    acc0 = __builtin_amdgcn_wmma_f32_16x16x4_f32(false, a, false, b0, (short)0, acc0, false, false);
    acc1 = __builtin_amdgcn_wmma_f32_16x16x4_f32(false, a, false, b1, (short)0, acc1, false, false);
    acc2 = __builtin_amdgcn_wmma_f32_16x16x4_f32(false, a, false, b2, (short)0, acc2, false, false);
    acc3 = __builtin_amdgcn_wmma_f32_16x16x4_f32(false, a, false, b3, (short)0, acc3, false, false);
  }

  // D layout: VGPR r holds element (M = r + 8*half, N = lm)
#pragma unroll
  for (int r = 0; r < 8; ++r) {
    float* __restrict__ Crow = C + (size_t)(row0 + r + 8 * half) * DD + col0 + lm;
    Crow[0]  = acc0[r];
    Crow[16] = acc1[r];
    Crow[32] = acc2[r];
    Crow[48] = acc3[r];
  }
}

// ---------------------------------------------------------------------------
// 3. edge aggregation: agg[dst] += h[src]  (one wave per edge, float4 per lane)
// ---------------------------------------------------------------------------
__global__ void __launch_bounds__(256) scatter_kernel(const float* __restrict__ h,
                                                      const int* __restrict__ src,
                                                      const int* __restrict__ dst,
                                                      float* __restrict__ agg) {
  const long long t = (long long)blockIdx.x * blockDim.x + threadIdx.x;
  const int e    = (int)(t >> 5);
  const int f4   = (int)(t & 31) * 4;
  if (e < EE) {
    const int s = src[e];
    const int d = dst[e];
    const float4 v = *(const float4*)(h + (size_t)s * DD + f4);
    float* __restrict__ o = agg + (size_t)d * DD + f4;
    atomicAdd(o + 0, v.x);
    atomicAdd(o + 1, v.y);
    atomicAdd(o + 2, v.z);
    atomicAdd(o + 3, v.w);
  }
}

// ---------------------------------------------------------------------------
// 4. in-place: agg = relu(agg * in_norm + b)
// ---------------------------------------------------------------------------
__global__ void __launch_bounds__(256) norm_bias_relu_kernel(float* __restrict__ agg,
                                                             const float* __restrict__ in_norm,
                                                             const float* __restrict__ b) {
  const int t  = blockIdx.x * blockDim.x + threadIdx.x; // < NN*32
  const int n  = t >> 5;
  const int f4 = (t & 31) * 4;
  if (n < NN) {
    const float s = in_norm[n];
    float4 v        = *(float4*)(agg + (size_t)n * DD + f4);
    const float4 bv = *(const float4*)(b + f4);
    v.x = fmaxf(v.x * s + bv.x, 0.0f);
    v.y = fmaxf(v.y * s + bv.y, 0.0f);
    v.z = fmaxf(v.z * s + bv.z, 0.0f);
    v.w = fmaxf(v.w * s + bv.w, 0.0f);
    *(float4*)(agg + (size_t)n * DD + f4) = v;
  }
}

// ---------------------------------------------------------------------------
// 5. final fused: h = agg*in_norm + b2 + res ; out = h @ mlpW + mlpb
//    One wave32 per node, butterfly reduction for the 2-wide head.
// ---------------------------------------------------------------------------
__global__ void __launch_bounds__(256) final_kernel(const float* __restrict__ agg,
                                                    const float* __restrict__ res,
                                                    const float* __restrict__ in_norm,
                                                    const float* __restrict__ b2,
                                                    const float* __restrict__ mlpW,
                                                    const float* __restrict__ mlpb,
                                                    float* __restrict__ out) {
  const int wave = threadIdx.x >> 5;
  const int lane = threadIdx.x & 31;
  const int n = blockIdx.x * 8 + wave;
  if (n >= NN) return;

  const float s = in_norm[n];
  const float* __restrict__ ag = agg + (size_t)n * DD;
  const float* __restrict__ rs = res + (size_t)n * DD;

  float a0 = 0.0f, a1 = 0.0f;
#pragma unroll
  for (int j = 0; j < 4; ++j) {
    const int f = lane + j * 32;
    const float hv = ag[f] * s + b2[f] + rs[f];
    a0 += hv * mlpW[f * 2 + 0];
    a1 += hv * mlpW[f * 2 + 1];
  }
#pragma unroll
  for (int off = 16; off > 0; off >>= 1) {
    a0 += __shfl_xor(a0, off, 32);
    a1 += __shfl_xor(a1, off, 32);
  }
  if (lane == 0) {
    out[n * 2 + 0] = a0 + mlpb[0];
    out[n * 2 + 1] = a1 + mlpb[1];
  }
}

// ---------------------------------------------------------------------------
extern "C" void kernel_launch(void* const* d_in, const int* in_sizes, int n_in,
                              void* d_out, int out_size, void* d_ws, size_t ws_size,
                              hipStream_t stream) {
  const float* x     = (const float*)d_in[0]; // [N,128]
  const int*   src   = (const int*)d_in[1];   // [E]
  const int*   dst   = (const int*)d_in[2];   // [E]
  const float* W1    = (const float*)d_in[3]; // [128,128]
  const float* b1    = (const float*)d_in[4]; // [128]
  const float* W2    = (const float*)d_in[5]; // [128,128]
  const float* b2    = (const float*)d_in[6]; // [128]
  const float* resW  = (const float*)d_in[7]; // [128,128]
  const float* mlpW  = (const float*)d_in[8]; // [128,2]
  const float* mlpb  = (const float*)d_in[9]; // [2]
  float* out = (float*)d_out;                 // [N,2]

  char* ws = (char*)d_ws;
  float* deg_src  = (float*)ws;          // N
  float* deg_dst  = deg_src + NN;        // N
  float* out_norm = deg_dst + NN;        // N
  float* in_norm  = out_norm + NN;       // N
  size_t off = ((size_t)4 * NN * sizeof(float) + 255) & ~(size_t)255;
  float* bufA = (float*)(ws + off);      // N*128 floats (51.2 MB)
  float* bufB = bufA + (size_t)NN * DD;  // N*128 floats (51.2 MB)

  const int thr = 256;

  // degrees + norms
  hipMemsetAsync(deg_src, 0, (size_t)2 * NN * sizeof(float), stream);
  deg_kernel<<<(EE + thr - 1) / thr, thr, 0, stream>>>(src, dst, deg_src, deg_dst);
  norm_kernel<<<(NN + thr - 1) / thr, thr, 0, stream>>>(deg_src, deg_dst, out_norm, in_norm);

  const int gemmGrid = (NN + 63) / 64;
  const long long scatThreads = (long long)EE * 32;
  const int scatGrid = (int)((scatThreads + thr - 1) / thr);
  const int elemGrid = (NN * 32 + thr - 1) / thr;

  // layer 1: bufA = (x * out_norm) @ W1 ; bufB = scatter ; bufB = relu(bufB*in_norm + b1)
  gemm_f32_wmma<<<gemmGrid, thr, 0, stream>>>(x, W1, out_norm, bufA, NN);
  hipMemsetAsync(bufB, 0, (size_t)NN * DD * sizeof(float), stream);
  scatter_kernel<<<scatGrid, thr, 0, stream>>>(bufA, src, dst, bufB);
  norm_bias_relu_kernel<<<elemGrid, thr, 0, stream>>>(bufB, in_norm, b1);

  // layer 2: bufA = (h1 * out_norm) @ W2 ; bufB = scatter
  gemm_f32_wmma<<<gemmGrid, thr, 0, stream>>>(bufB, W2, out_norm, bufA, NN);
  hipMemsetAsync(bufB, 0, (size_t)NN * DD * sizeof(float), stream); // stream-ordered after GEMM read
  scatter_kernel<<<scatGrid, thr, 0, stream>>>(bufA, src, dst, bufB);

  // residual: bufA = x @ res_W (no row scale) — stream-ordered after scatter read of bufA
  gemm_f32_wmma<<<gemmGrid, thr, 0, stream>>>(x, resW, nullptr, bufA, NN);

  // final fused: out = (bufB*in_norm + b2 + bufA) @ mlpW + mlpb
  final_kernel<<<(NN + 7) / 8, thr, 0, stream>>>(bufB, bufA, in_norm, b2, mlpW, mlpb, out);
}